// BuildingGenModel_7533372637726
// MI455X (gfx1250) — compile-verified
//
#include <hip/hip_runtime.h>
#include <math.h>

typedef __attribute__((ext_vector_type(2))) float v2f;
typedef __attribute__((ext_vector_type(8))) float v8f;

// ---------------------------------------------------------------------------
// float <-> order-preserving u32 encoding: segment-min becomes deterministic
// GLOBAL_ATOMIC_MIN_U32 (order-independent => bitwise reproducible).
// ---------------------------------------------------------------------------
__device__ __forceinline__ unsigned enc_f32(float f) {
    unsigned u = __float_as_uint(f);
    return (u & 0x80000000u) ? ~u : (u | 0x80000000u);
}
// enc(FLT_MAX) == 0xFF7FFFFF; anything >= that (incl. untouched 0xFFFFFFFF)
// means "no in-edge" -> 0.0f per the reference's where(agg >= finfo.max, 0).
__device__ __forceinline__ float dec_f32(unsigned e) {
    unsigned u = (e & 0x80000000u) ? (e & 0x7FFFFFFFu) : ~e;
    float f = __uint_as_float(u);
    return (e >= 0xFF7FFFFFu) ? 0.0f : f;
}

// ---------------------------------------------------------------------------
// segment-min over edges: one thread per (edge, feature); D compile-time so
// the div/mod strength-reduce to shifts.
// ---------------------------------------------------------------------------
template <int D>
__global__ void scatter_min_kernel(const float* __restrict__ x,
                                   const int* __restrict__ src,
                                   const int* __restrict__ dst,
                                   unsigned* __restrict__ agg,
                                   unsigned total) {
    unsigned idx = blockIdx.x * blockDim.x + threadIdx.x;
    if (idx >= total) return;
    unsigned e = idx / (unsigned)D;
    unsigned f = idx - e * (unsigned)D;
    int s = src[e];
    int t = dst[e];
    unsigned v = enc_f32(x[(size_t)s * D + f]);
    atomicMin(&agg[(size_t)t * D + f], v);
}

// ---------------------------------------------------------------------------
// Fused SAGE linear:  y = dec(agg) @ wl + x @ wr + bl   (fp32 WMMA 16x16x4)
// One wave -> 16(M) x (NT*16)(N) tile; each A pair feeds 2*NT WMMAs.
// A 16x4 layout: lane%16 = M, K = 2*(lane/16) + vgpr.
// B 4x16 layout: lane%16 = N, K = 2*(lane/16) + vgpr.
// C/D: vgpr r -> M = r + 8*(lane/16), N = lane%16.
// All dims padded so there are NO guards in the K loop.
// blockDim = 128 (4 waves, 4 m-tiles); grid = (ceil(n/64), COUT/(NT*16)).
// ---------------------------------------------------------------------------
template <int CIN, int COUT, int NT>
__global__ void sage_gemm_kernel(const unsigned* __restrict__ agg,
                                 const float* __restrict__ xin,
                                 const float* __restrict__ wl,
                                 const float* __restrict__ wr,
                                 const float* __restrict__ bl,
                                 float* __restrict__ y,
                                 int n) {
    const int lane  = threadIdx.x & 31;
    const int wave  = threadIdx.x >> 5;
    const int mtile = blockIdx.x * 4 + wave;
    if (mtile * 16 >= n) return;              // wave-uniform: EXEC all-1s
    const int lm    = lane & 15;
    const int hi    = lane >> 4;              // 0 or 1
    const int khalf = hi << 1;                // 0 or 2
    const int ncb   = blockIdx.y * (NT * 16) + lm;
    int arow = mtile * 16 + lm;
    if (arow >= n) arow = n - 1;              // only possible if n % 16 != 0

    const unsigned* ag = agg + (size_t)arow * CIN;
    const float*    xr = xin + (size_t)arow * CIN;

    v8f acc[NT];
#pragma unroll
    for (int t = 0; t < NT; ++t) acc[t] = (v8f){};

#pragma unroll 2
    for (int k0 = 0; k0 < CIN; k0 += 4) {
        const int ka = k0 + khalf;            // even => 8B-aligned pair loads
        uint2 e = *(const uint2*)(ag + ka);
        v2f aA, aX;
        aA.x = dec_f32(e.x);
        aA.y = dec_f32(e.y);
        aX = *(const v2f*)(xr + ka);
        v2f bL[NT], bR[NT];
#pragma unroll
        for (int t = 0; t < NT; ++t) {
            const int nc = ncb + t * 16;
            bL[t].x = wl[(size_t)ka * COUT + nc];
            bL[t].y = wl[(size_t)(ka + 1) * COUT + nc];
            bR[t].x = wr[(size_t)ka * COUT + nc];
            bR[t].y = wr[(size_t)(ka + 1) * COUT + nc];
        }
#pragma unroll
        for (int t = 0; t < NT; ++t)
            acc[t] = __builtin_amdgcn_wmma_f32_16x16x4_f32(
                false, aA, false, bL[t], (short)0, acc[t], false, false);
#pragma unroll
        for (int t = 0; t < NT; ++t)
            acc[t] = __builtin_amdgcn_wmma_f32_16x16x4_f32(
                false, aX, false, bR[t], (short)0, acc[t], false, false);
    }

    const int mh = hi * 8;
#pragma unroll
    for (int t = 0; t < NT; ++t) {
        const int nc = ncb + t * 16;
        const float bias = bl[nc];
#pragma unroll
        for (int r = 0; r < 8; ++r) {
            const int m = mtile * 16 + mh + r;
            if (m < n) y[(size_t)m * COUT + nc] = acc[t][r] + bias;
        }
    }
}

// ---------------------------------------------------------------------------
// Deterministic BatchNorm statistics: per-block partials, fixed-order reduce.
// ---------------------------------------------------------------------------
__global__ void bn_stats_partial_kernel(const float* __restrict__ y,
                                        float* __restrict__ partial,
                                        int n, int d) {
    __shared__ float sd[512];
    const int col = threadIdx.x % d;
    const int rg  = threadIdx.x / d;
    const int rpb = blockDim.x / d;
    float s = 0.0f, s2 = 0.0f;
    for (int r = blockIdx.x * rpb + rg; r < n; r += gridDim.x * rpb) {
        float v = y[(size_t)r * d + col];
        s += v; s2 += v * v;
    }
    sd[threadIdx.x] = s;
    sd[256 + threadIdx.x] = s2;
    __syncthreads();
    if (rg == 0) {
        for (int j = 1; j < rpb; ++j) {
            s  += sd[j * d + col];
            s2 += sd[256 + j * d + col];
        }
        partial[(size_t)blockIdx.x * 2 * d + col]     = s;
        partial[(size_t)blockIdx.x * 2 * d + d + col] = s2;
    }
}

__global__ void bn_stats_reduce_kernel(const float* __restrict__ partial,
                                       float* __restrict__ stats,
                                       int nblocks, int d) {
    int i = blockIdx.x * blockDim.x + threadIdx.x;
    if (i >= 2 * d) return;
    float s = 0.0f;
    for (int b = 0; b < nblocks; ++b) s += partial[(size_t)b * 2 * d + i];
    stats[i] = s;
}

template <int D>
__global__ void bn_apply_relu_kernel(float* __restrict__ y,
                                     const float* __restrict__ stats,
                                     const float* __restrict__ g,
                                     const float* __restrict__ b,
                                     int n) {
    unsigned idx = blockIdx.x * blockDim.x + threadIdx.x;
    unsigned total = (unsigned)n * (unsigned)D;
    if (idx >= total) return;
    int col = idx % (unsigned)D;
    float inv_n = 1.0f / (float)n;
    float mu  = stats[col] * inv_n;
    float var = stats[D + col] * inv_n - mu * mu;
    float v = (y[idx] - mu) * rsqrtf(var + 1e-5f) * g[col] + b[col];
    y[idx] = v > 0.0f ? v : 0.0f;
}

// ---------------------------------------------------------------------------
// padded concat: xs[n,36] = [x (32) | rm (3, stride 16) | 0]
// ---------------------------------------------------------------------------
__global__ void concat_kernel(const float* __restrict__ x,
                              const float* __restrict__ rm,
                              float* __restrict__ xs, int n) {
    unsigned idx = blockIdx.x * blockDim.x + threadIdx.x;
    unsigned total = (unsigned)n * 36u;
    if (idx >= total) return;
    unsigned i = idx / 36u, f = idx - i * 36u;
    float v;
    if (f < 32u)      v = x[(size_t)i * 32 + f];
    else if (f < 35u) v = rm[(size_t)i * 16 + (f - 32u)];
    else              v = 0.0f;
    xs[idx] = v;
}

// weight padding: rm4 [64,3]->[64,16] (+bias), sh1 [35,128]->[36,128]
__global__ void pad_rm4_kernel(const float* __restrict__ wl,
                               const float* __restrict__ wr,
                               const float* __restrict__ bl,
                               float* __restrict__ wlp,
                               float* __restrict__ wrp,
                               float* __restrict__ blp) {
    int i = blockIdx.x * blockDim.x + threadIdx.x;
    if (i < 16) blp[i] = (i < 3) ? bl[i] : 0.0f;
    if (i >= 64 * 16) return;
    int k = i >> 4, j = i & 15;
    float l = (j < 3) ? wl[k * 3 + j] : 0.0f;
    float r = (j < 3) ? wr[k * 3 + j] : 0.0f;
    wlp[i] = l;
    wrp[i] = r;
}

__global__ void pad_sh1_kernel(const float* __restrict__ wl,
                               const float* __restrict__ wr,
                               float* __restrict__ wlp,
                               float* __restrict__ wrp) {
    int i = blockIdx.x * blockDim.x + threadIdx.x;
    if (i >= 36 * 128) return;
    float l = (i < 35 * 128) ? wl[i] : 0.0f;
    float r = (i < 35 * 128) ? wr[i] : 0.0f;
    wlp[i] = l;
    wrp[i] = r;
}

// log_softmax over 3 classes (rm stride 16) + labels = (argmax == 2)
__global__ void softmax_labels_kernel(const float* __restrict__ rm,
                                      float* __restrict__ out_ls,
                                      float* __restrict__ labels, int n) {
    int i = blockIdx.x * blockDim.x + threadIdx.x;
    if (i >= n) return;
    float a0 = rm[(size_t)i * 16 + 0];
    float a1 = rm[(size_t)i * 16 + 1];
    float a2 = rm[(size_t)i * 16 + 2];
    float m = fmaxf(a0, fmaxf(a1, a2));
    float lse = logf(expf(a0 - m) + expf(a1 - m) + expf(a2 - m)) + m;
    out_ls[(size_t)i * 3 + 0] = a0 - lse;
    out_ls[(size_t)i * 3 + 1] = a1 - lse;
    out_ls[(size_t)i * 3 + 2] = a2 - lse;
    int am = 0; float best = a0;               // first-max tie-break = jnp.argmax
    if (a1 > best) { best = a1; am = 1; }
    if (a2 > best) { am = 2; }
    labels[i] = (am == 2) ? 1.0f : 0.0f;
}

// out[i] = (feat[i,:64] . w + b) * labels[i]   (one wave32 per node)
__global__ void head_kernel(const float* __restrict__ feat,
                            const float* __restrict__ w,
                            const float* __restrict__ b,
                            const float* __restrict__ labels,
                            float* __restrict__ out, int n) {
    int lane = threadIdx.x & 31;
    int node = (blockIdx.x * blockDim.x + threadIdx.x) >> 5;
    if (node >= n) return;
    float s = feat[(size_t)node * 64 + lane]      * w[lane]
            + feat[(size_t)node * 64 + 32 + lane] * w[32 + lane];
#pragma unroll
    for (int o = 16; o > 0; o >>= 1) s += __shfl_xor(s, o, 32);
    if (lane == 0) out[node] = (s + b[0]) * labels[node];
}

// ---------------------------------------------------------------------------
// host side
// ---------------------------------------------------------------------------
extern "C" void kernel_launch(void* const* d_in, const int* in_sizes, int n_in,
                              void* d_out, int out_size, void* d_ws, size_t ws_size,
                              hipStream_t stream) {
    const int C = 32;
    const int n = in_sizes[0] / C;
    const int E = in_sizes[1] / 2;

    const float* x   = (const float*)d_in[0];
    const int*   ei  = (const int*)d_in[1];
    const int*   src = ei;
    const int*   dst = ei + E;

    const float *rm1_wl=(const float*)d_in[2],  *rm1_bl=(const float*)d_in[3],
                *rm1_wr=(const float*)d_in[4],  *rmn1_g=(const float*)d_in[5],
                *rmn1_b=(const float*)d_in[6];
    const float *rm2_wl=(const float*)d_in[7],  *rm2_bl=(const float*)d_in[8],
                *rm2_wr=(const float*)d_in[9],  *rmn2_g=(const float*)d_in[10],
                *rmn2_b=(const float*)d_in[11];
    const float *rm4_wl=(const float*)d_in[12], *rm4_bl=(const float*)d_in[13],
                *rm4_wr=(const float*)d_in[14];
    const float *sh1_wl=(const float*)d_in[15], *sh1_bl=(const float*)d_in[16],
                *sh1_wr=(const float*)d_in[17], *shn1_g=(const float*)d_in[18],
                *shn1_b=(const float*)d_in[19];
    const float *rt1_wl=(const float*)d_in[20], *rt1_bl=(const float*)d_in[21],
                *rt1_wr=(const float*)d_in[22], *rtn1_g=(const float*)d_in[23],
                *rtn1_b=(const float*)d_in[24], *rt3_w=(const float*)d_in[25],
                *rt3_b=(const float*)d_in[26];
    const float *md1_wl=(const float*)d_in[27], *md1_bl=(const float*)d_in[28],
                *md1_wr=(const float*)d_in[29], *mdn1_g=(const float*)d_in[30],
                *mdn1_b=(const float*)d_in[31], *md3_w=(const float*)d_in[32],
                *md3_b=(const float*)d_in[33];

    // workspace carve (256B aligned)
    char* ws = (char*)d_ws;
    size_t off = 0;
    auto carve = [&](size_t bytes) -> void* {
        void* p = ws + off;
        off += (bytes + 255) & ~(size_t)255;
        return p;
    };
    const int NSTAT_BLK = 256;
    float*    A    = (float*)carve((size_t)n * 128 * 4);   // rm1 out, then sh1 out
    float*    B    = (float*)carve((size_t)n * 64 * 4);    // rm2, then rt1/md1 out
    float*    RM   = (float*)carve((size_t)n * 16 * 4);    // rm4 out (padded 3->16)
    float*    XS   = (float*)carve((size_t)n * 36 * 4);    // concat (padded 35->36)
    unsigned* AGG  = (unsigned*)carve((size_t)n * 128 * 4);
    float*    LB   = (float*)carve((size_t)n * 4);
    float*    PART = (float*)carve((size_t)NSTAT_BLK * 256 * 4);
    float*    ST   = (float*)carve(256 * 4);
    float*    WLP4 = (float*)carve(64 * 16 * 4);           // rm4 wl padded
    float*    WRP4 = (float*)carve(64 * 16 * 4);           // rm4 wr padded
    float*    BLP4 = (float*)carve(16 * 4);                // rm4 bl padded
    float*    WLPS = (float*)carve(36 * 128 * 4);          // sh1 wl padded
    float*    WRPS = (float*)carve(36 * 128 * 4);          // sh1 wr padded

    const int mblk = ( (n + 15) / 16 + 3 ) / 4;            // 4 m-tiles per block

    auto bn = [&](float* y, const float* g, const float* b, int d) {
        bn_stats_partial_kernel<<<NSTAT_BLK, 256, 0, stream>>>(y, PART, n, d);
        bn_stats_reduce_kernel<<<1, 256, 0, stream>>>(PART, ST, NSTAT_BLK, d);
        unsigned total = (unsigned)n * (unsigned)d;
        if (d == 128)
            bn_apply_relu_kernel<128><<<(total + 255) / 256, 256, 0, stream>>>(y, ST, g, b, n);
        else
            bn_apply_relu_kernel<64><<<(total + 255) / 256, 256, 0, stream>>>(y, ST, g, b, n);
    };

    // weight padding (tiny, once per call)
    pad_rm4_kernel<<<(64 * 16 + 255) / 256, 256, 0, stream>>>(rm4_wl, rm4_wr, rm4_bl,
                                                              WLP4, WRP4, BLP4);
    pad_sh1_kernel<<<(36 * 128 + 255) / 256, 256, 0, stream>>>(sh1_wl, sh1_wr, WLPS, WRPS);

    // ---- rm1: SAGE(32 -> 128) + BN + ReLU ----
    hipMemsetAsync(AGG, 0xFF, (size_t)n * 32 * 4, stream);
    {
        unsigned total = (unsigned)E * 32u;
        scatter_min_kernel<32><<<(total + 255) / 256, 256, 0, stream>>>(x, src, dst, AGG, total);
    }
    sage_gemm_kernel<32, 128, 4><<<dim3(mblk, 2), 128, 0, stream>>>(
        AGG, x, rm1_wl, rm1_wr, rm1_bl, A, n);
    bn(A, rmn1_g, rmn1_b, 128);

    // ---- rm2: SAGE(128 -> 64) + BN + ReLU ----
    hipMemsetAsync(AGG, 0xFF, (size_t)n * 128 * 4, stream);
    {
        unsigned total = (unsigned)E * 128u;
        scatter_min_kernel<128><<<(total + 255) / 256, 256, 0, stream>>>(A, src, dst, AGG, total);
    }
    sage_gemm_kernel<128, 64, 4><<<dim3(mblk, 1), 128, 0, stream>>>(
        AGG, A, rm2_wl, rm2_wr, rm2_bl, B, n);
    bn(B, rmn2_g, rmn2_b, 64);

    // ---- rm4: SAGE(64 -> 3 padded to 16) ----
    hipMemsetAsync(AGG, 0xFF, (size_t)n * 64 * 4, stream);
    {
        unsigned total = (unsigned)E * 64u;
        scatter_min_kernel<64><<<(total + 255) / 256, 256, 0, stream>>>(B, src, dst, AGG, total);
    }
    sage_gemm_kernel<64, 16, 1><<<dim3(mblk, 1), 128, 0, stream>>>(
        AGG, B, WLP4, WRP4, BLP4, RM, n);

    // ---- xs = concat([x, rm]) padded to 36 ----
    {
        unsigned total = (unsigned)n * 36u;
        concat_kernel<<<(total + 255) / 256, 256, 0, stream>>>(x, RM, XS, n);
    }

    // ---- sh1: SAGE(36 -> 128) + BN + ReLU (result reuses A) ----
    hipMemsetAsync(AGG, 0xFF, (size_t)n * 36 * 4, stream);
    {
        unsigned total = (unsigned)E * 36u;
        scatter_min_kernel<36><<<(total + 255) / 256, 256, 0, stream>>>(XS, src, dst, AGG, total);
    }
    sage_gemm_kernel<36, 128, 4><<<dim3(mblk, 2), 128, 0, stream>>>(
        AGG, XS, WLPS, WRPS, sh1_bl, A, n);
    bn(A, shn1_g, shn1_b, 128);

    // ---- rm_ls + labels ----
    float* out = (float*)d_out;
    softmax_labels_kernel<<<(n + 255) / 256, 256, 0, stream>>>(RM, out, LB, n);

    // ---- rt1: SAGE(128 -> 64) + BN + ReLU, head -> out[3n..4n) ----
    hipMemsetAsync(AGG, 0xFF, (size_t)n * 128 * 4, stream);
    {
        unsigned total = (unsigned)E * 128u;
        scatter_min_kernel<128><<<(total + 255) / 256, 256, 0, stream>>>(A, src, dst, AGG, total);
    }
    sage_gemm_kernel<128, 64, 4><<<dim3(mblk, 1), 128, 0, stream>>>(
        AGG, A, rt1_wl, rt1_wr, rt1_bl, B, n);
    bn(B, rtn1_g, rtn1_b, 64);
    head_kernel<<<((size_t)n * 32 + 255) / 256, 256, 0, stream>>>(
        B, rt3_w, rt3_b, LB, out + (size_t)3 * n, n);

    // ---- md1: same aggregation as rt1 -> reuse AGG (skip scatter) ----
    sage_gemm_kernel<128, 64, 4><<<dim3(mblk, 1), 128, 0, stream>>>(
        AGG, A, md1_wl, md1_wr, md1_bl, B, n);
    bn(B, mdn1_g, mdn1_b, 64);
    head_kernel<<<((size_t)n * 32 + 255) / 256, 256, 0, stream>>>(
        B, md3_w, md3_b, LB, out + (size_t)4 * n, n);
}